// SSDFaceLoss_26731876451093
// MI455X (gfx1250) — compile-verified
//
#include <hip/hip_runtime.h>
#include <cstdint>

#define B_ 32
#define P_ 131072
#define M_ 16
#define NEG_POS_RATE_ 3

// ---------------- workspace layout ----------------
// bytes [0, 4096): bestPack  (B*M u64, packed (iou_bits<<32)|~prior_idx)
// bytes [4096, ...): uint32 arena, word offsets below
static constexpr size_t OFF_U32 = 4096;
static constexpr int W_HIST0   = 0;                     // B*4096
static constexpr int W_HIST1   = W_HIST0 + B_ * 4096;   // B*4096
static constexpr int W_HIST2   = W_HIST1 + B_ * 4096;   // B*256
static constexpr int W_POSCNT  = W_HIST2 + B_ * 256;    // B
static constexpr int W_PREHI   = W_POSCNT + B_;         // B
static constexpr int W_PREMID  = W_PREHI + B_;          // B
static constexpr int W_TBITS   = W_PREMID + B_;         // B
static constexpr int W_CNTGT   = W_TBITS + B_;          // B
static constexpr int W_KTGT    = W_CNTGT + B_;          // B
static constexpr int W_BLKLOC  = W_KTGT + B_;           // B*128 floats
static constexpr int W_BLKPC   = W_BLKLOC + B_ * 128;   // B*128 floats
static constexpr int W_BLKNEG  = W_BLKPC + B_ * 128;    // B*32 floats
static constexpr int W_END     = W_BLKNEG + B_ * 32;
static constexpr int W_NEGLOSS = ((W_END + 15) / 16) * 16;  // B*P floats
static constexpr int W_ZEROCNT = 1024 + W_NEGLOSS;          // words to zero (incl bestPack)

typedef __attribute__((ext_vector_type(2))) float v2f;
typedef __attribute__((ext_vector_type(8))) float v8f;

// ---------------- helpers ----------------
__device__ __forceinline__ float softplusf(float x) {
  return fmaxf(x, 0.0f) + log1pf(expf(-fabsf(x)));
}
__device__ __forceinline__ float smoothl1(float d) {
  float ad = fabsf(d);
  return (ad < 1.0f) ? 0.5f * d * d : (ad - 0.5f);
}

// Exact 32-lane f32 sum via V_WMMA_F32_16X16X4_F32 (all lanes receive total).
__device__ __forceinline__ float wave_reduce_sum(float v) {
#if __has_builtin(__builtin_amdgcn_wmma_f32_16x16x4_f32)
  v2f a;    a.x = v;    a.y = 0.0f;
  v2f ones; ones.x = 1.0f; ones.y = 1.0f;
  v8f z = {};
  // D1[m][n] = v[m] + v[m+16]  (A-layout: lane<16 holds K0,K1; lane>=16 holds K2,K3)
  v8f d1 = __builtin_amdgcn_wmma_f32_16x16x4_f32(false, a, false, ones, (short)0, z,
                                                 false, false);
  // Sum the 16 row-values via ones-A, chaining C across 4 K=4 steps.
  v8f acc = {};
  v2f b;
  b.x = d1[0]; b.y = d1[1];
  acc = __builtin_amdgcn_wmma_f32_16x16x4_f32(false, ones, false, b, (short)0, acc, false, false);
  b.x = d1[2]; b.y = d1[3];
  acc = __builtin_amdgcn_wmma_f32_16x16x4_f32(false, ones, false, b, (short)0, acc, false, false);
  b.x = d1[4]; b.y = d1[5];
  acc = __builtin_amdgcn_wmma_f32_16x16x4_f32(false, ones, false, b, (short)0, acc, false, false);
  b.x = d1[6]; b.y = d1[7];
  acc = __builtin_amdgcn_wmma_f32_16x16x4_f32(false, ones, false, b, (short)0, acc, false, false);
  return acc[0];
#else
  for (int off = 16; off > 0; off >>= 1) v += __shfl_xor(v, off, 32);
  return v;
#endif
}

// ---------------- kernels ----------------
__global__ __launch_bounds__(256) void k_init(char* __restrict__ ws) {
  uint32_t* w = (uint32_t*)ws;
  int idx = blockIdx.x * 256 + threadIdx.x;
  int stride = gridDim.x * 256;
  for (int i = idx; i < W_ZEROCNT; i += stride) w[i] = 0u;
}

// Per-truth best prior (argmax over axis=1, first-max tie-break via ~p).
__global__ __launch_bounds__(256) void k_best_prior(const float* __restrict__ priors,
                                                    const float* __restrict__ targets,
                                                    char* __restrict__ ws) {
  unsigned long long* bestPack = (unsigned long long*)ws;
  const int b = blockIdx.x >> 6;
  const int blk = blockIdx.x & 63;
  const int tid = threadIdx.x;

  __shared__ float tr[M_][4];
  __shared__ float trArea[M_];
  __shared__ unsigned long long sBest[M_];
  if (tid < M_ * 5) {
    int m = tid / 5, c = tid % 5;
    if (c < 4) tr[m][c] = targets[(b * M_ + m) * 5 + c];
  }
  if (tid < M_) sBest[tid] = 0ull;
  __syncthreads();
  if (tid < M_) trArea[tid] = (tr[tid][2] - tr[tid][0]) * (tr[tid][3] - tr[tid][1]);
  __syncthreads();

  unsigned long long best[M_];
#pragma unroll
  for (int m = 0; m < M_; ++m) best[m] = 0ull;

  const int base = blk * 2048;
  for (int i = 0; i < 8; ++i) {
    int p = base + i * 256 + tid;
    float4 pr = ((const float4*)priors)[p];
    float bx0 = pr.x - 0.5f * pr.z, by0 = pr.y - 0.5f * pr.w;
    float bx1 = pr.x + 0.5f * pr.z, by1 = pr.y + 0.5f * pr.w;
    float areaB = pr.z * pr.w;
    unsigned int lowkey = ~(unsigned int)p;
#pragma unroll
    for (int m = 0; m < M_; ++m) {
      float lt0 = fmaxf(tr[m][0], bx0), lt1 = fmaxf(tr[m][1], by0);
      float rb0 = fminf(tr[m][2], bx1), rb1 = fminf(tr[m][3], by1);
      float w = fmaxf(rb0 - lt0, 0.0f), h = fmaxf(rb1 - lt1, 0.0f);
      float inter = w * h;
      float iou = inter / (trArea[m] + areaB - inter);
      unsigned long long key =
          ((unsigned long long)__float_as_uint(iou) << 32) | (unsigned long long)lowkey;
      best[m] = (key > best[m]) ? key : best[m];
    }
  }
#pragma unroll
  for (int m = 0; m < M_; ++m) atomicMax(&sBest[m], best[m]);
  __syncthreads();
  if (tid < M_) atomicMax(&bestPack[b * M_ + tid], sBest[tid]);
}

// Main pass: matching, loc loss, pos BCE, neg_loss array + level-0 histogram.
__global__ __launch_bounds__(256) void k_main(const float* __restrict__ loc_p,
                                              const float* __restrict__ conf_p,
                                              const float* __restrict__ priors,
                                              const float* __restrict__ targets,
                                              char* __restrict__ ws) {
  const unsigned long long* bestPack = (const unsigned long long*)ws;
  uint32_t* u = (uint32_t*)(ws + OFF_U32);
  uint32_t* hist0 = u + W_HIST0;
  uint32_t* posCnt = u + W_POSCNT;
  float* blkLoc = (float*)(u + W_BLKLOC);
  float* blkPC = (float*)(u + W_BLKPC);
  float* negLoss = (float*)(u + W_NEGLOSS);

  const int b = blockIdx.x >> 7;
  const int blk = blockIdx.x & 127;
  const int tid = threadIdx.x;

  __shared__ float tr[M_][4];
  __shared__ float trArea[M_];
  __shared__ int bpIdx[M_];
  __shared__ uint32_t sh[4096];
  __shared__ float wLoc[8], wPC[8], wCnt[8];

  if (tid < M_ * 5) {
    int m = tid / 5, c = tid % 5;
    if (c < 4) tr[m][c] = targets[(b * M_ + m) * 5 + c];
  }
  if (tid < M_) bpIdx[tid] = (int)(~(unsigned int)(bestPack[b * M_ + tid] & 0xFFFFFFFFull));
  for (int j = tid; j < 4096; j += 256) sh[j] = 0u;
  __syncthreads();
  if (tid < M_) trArea[tid] = (tr[tid][2] - tr[tid][0]) * (tr[tid][3] - tr[tid][1]);
  __syncthreads();

  float accLoc = 0.0f, accPC = 0.0f, accCnt = 0.0f;
  for (int i = 0; i < 4; ++i) {
    int p = blk * 1024 + i * 256 + tid;
    int gp = b * P_ + p;
    float4 pr = ((const float4*)priors)[p];
    float bx0 = pr.x - 0.5f * pr.z, by0 = pr.y - 0.5f * pr.w;
    float bx1 = pr.x + 0.5f * pr.z, by1 = pr.y + 0.5f * pr.w;
    float areaB = pr.z * pr.w;
    float bestIou = -1.0f;
    int bestM = 0;
#pragma unroll
    for (int m = 0; m < M_; ++m) {
      float lt0 = fmaxf(tr[m][0], bx0), lt1 = fmaxf(tr[m][1], by0);
      float rb0 = fminf(tr[m][2], bx1), rb1 = fminf(tr[m][3], by1);
      float w = fmaxf(rb0 - lt0, 0.0f), h = fmaxf(rb1 - lt1, 0.0f);
      float inter = w * h;
      float iou = inter / (trArea[m] + areaB - inter);
      if (iou > bestIou) { bestIou = iou; bestM = m; }
    }
    bool forced = false;
#pragma unroll
    for (int m = 0; m < M_; ++m)
      if (bpIdx[m] == p) { bestM = m; forced = true; }  // last-write-wins like .at[].set
    bool pos = forced || (bestIou >= 0.5f);
    float cv = conf_p[gp];
    float nl;
    if (pos) {
      float t0 = tr[bestM][0], t1 = tr[bestM][1], t2 = tr[bestM][2], t3 = tr[bestM][3];
      float g0 = ((t0 + t2) * 0.5f - pr.x) / (0.1f * pr.z);
      float g1 = ((t1 + t3) * 0.5f - pr.y) / (0.1f * pr.w);
      float g2 = logf((t2 - t0) / pr.z) * 5.0f;
      float g3 = logf((t3 - t1) / pr.w) * 5.0f;
      float4 lp = ((const float4*)loc_p)[gp];
      accLoc += smoothl1(lp.x - g0) + smoothl1(lp.y - g1) + smoothl1(lp.z - g2) +
                smoothl1(lp.w - g3);
      accPC += softplusf(-cv);
      accCnt += 1.0f;
      nl = 0.0f;
    } else {
      nl = softplusf(cv);
    }
    negLoss[gp] = nl;
    atomicAdd(&sh[__float_as_uint(nl) >> 20], 1u);
  }

  // per-wave exact sums via WMMA (EXEC all ones here: uniform control flow)
  float sLoc = wave_reduce_sum(accLoc);
  float sPC = wave_reduce_sum(accPC);
  float sCnt = wave_reduce_sum(accCnt);
  if ((tid & 31) == 0) {
    wLoc[tid >> 5] = sLoc;
    wPC[tid >> 5] = sPC;
    wCnt[tid >> 5] = sCnt;
  }
  __syncthreads();
  for (int j = tid; j < 4096; j += 256) {
    uint32_t c = sh[j];
    if (c) atomicAdd(&hist0[b * 4096 + j], c);
  }
  if (tid == 0) {
    float tl = 0.0f, tp = 0.0f, tc = 0.0f;
    for (int wv = 0; wv < 8; ++wv) { tl += wLoc[wv]; tp += wPC[wv]; tc += wCnt[wv]; }
    blkLoc[b * 128 + blk] = tl;
    blkPC[b * 128 + blk] = tp;
    atomicAdd(&posCnt[b], (uint32_t)(tc + 0.5f));
  }
}

// Radix-select step: pick bucket containing k-th largest, accumulate strict-greater count.
__global__ void k_select(char* __restrict__ ws, int level) {
  if (threadIdx.x != 0) return;
  uint32_t* u = (uint32_t*)(ws + OFF_U32);
  const int b = blockIdx.x;
  uint32_t* posCnt = u + W_POSCNT;
  uint32_t* preHi = u + W_PREHI;
  uint32_t* preMid = u + W_PREMID;
  uint32_t* tBits = u + W_TBITS;
  uint32_t* cntGt = u + W_CNTGT;
  uint32_t* kTgt = u + W_KTGT;

  if (level == 0) {
    unsigned long long k64 = (unsigned long long)posCnt[b] * NEG_POS_RATE_;
    uint32_t k = (k64 > (unsigned long long)P_) ? (uint32_t)P_ : (uint32_t)k64;
    kTgt[b] = k;
    if (k == 0u) {
      preHi[b] = 0x1000u;  // matches no 12-bit prefix
      preMid[b] = 0x1000u;
      tBits[b] = 0xFFFFFFFFu;  // nothing strictly greater
      cntGt[b] = 0u;
      return;
    }
    const uint32_t* h = u + W_HIST0 + b * 4096;
    uint32_t cum = 0;
    for (int i = 4095; i >= 0; --i) {
      uint32_t c = h[i];
      if (cum + c >= k) { preHi[b] = (uint32_t)i; cntGt[b] = cum; return; }
      cum += c;
    }
  } else if (level == 1) {
    if (kTgt[b] == 0u) return;
    uint32_t kr = kTgt[b] - cntGt[b];
    const uint32_t* h = u + W_HIST1 + b * 4096;
    uint32_t cum = 0;
    for (int i = 4095; i >= 0; --i) {
      uint32_t c = h[i];
      if (cum + c >= kr) { preMid[b] = (uint32_t)i; cntGt[b] += cum; return; }
      cum += c;
    }
  } else {
    if (kTgt[b] == 0u) return;
    uint32_t kr = kTgt[b] - cntGt[b];
    const uint32_t* h = u + W_HIST2 + b * 256;
    uint32_t cum = 0;
    for (int i = 255; i >= 0; --i) {
      uint32_t c = h[i];
      if (cum + c >= kr) {
        tBits[b] = (preHi[b] << 20) | (preMid[b] << 8) | (uint32_t)i;
        cntGt[b] += cum;
        return;
      }
      cum += c;
    }
  }
}

// Refinement histograms (level 1: mid 12 bits; level 2: low 8 bits).
__global__ __launch_bounds__(256) void k_hist_refine(char* __restrict__ ws, int level) {
  uint32_t* u = (uint32_t*)(ws + OFF_U32);
  const float* negLoss = (const float*)(u + W_NEGLOSS);
  const int b = blockIdx.x >> 5;
  const int blk = blockIdx.x & 31;
  const int tid = threadIdx.x;

  __shared__ uint32_t sh[4096];
  const int nb = (level == 1) ? 4096 : 256;
  for (int j = tid; j < nb; j += 256) sh[j] = 0u;
  __syncthreads();

  uint32_t preHi = u[W_PREHI + b];
  uint32_t preMid = u[W_PREMID + b];
  uint32_t pref24 = (preHi << 12) | preMid;
  const int base = b * P_ + blk * 4096;
  for (int i = 0; i < 16; ++i) {
    uint32_t key = __float_as_uint(negLoss[base + i * 256 + tid]);
    if (level == 1) {
      if ((key >> 20) == preHi) atomicAdd(&sh[(key >> 8) & 0xFFFu], 1u);
    } else {
      if ((key >> 8) == pref24) atomicAdd(&sh[key & 0xFFu], 1u);
    }
  }
  __syncthreads();
  uint32_t* ho = (level == 1) ? (u + W_HIST1 + b * 4096) : (u + W_HIST2 + b * 256);
  for (int j = tid; j < nb; j += 256) {
    uint32_t c = sh[j];
    if (c) atomicAdd(&ho[j], c);
  }
}

// Sum of neg_loss values strictly greater than threshold, per batch (block partials).
__global__ __launch_bounds__(256) void k_sum_gt(char* __restrict__ ws) {
  uint32_t* u = (uint32_t*)(ws + OFF_U32);
  const float* negLoss = (const float*)(u + W_NEGLOSS);
  float* blkNeg = (float*)(u + W_BLKNEG);
  const int b = blockIdx.x >> 5;
  const int blk = blockIdx.x & 31;
  const int tid = threadIdx.x;
  const uint32_t t = u[W_TBITS + b];

  float acc = 0.0f;
  const int base = b * P_ + blk * 4096;
  for (int i = 0; i < 16; ++i) {
    float v = negLoss[base + i * 256 + tid];
    if (__float_as_uint(v) > t) acc += v;
  }
  float s = wave_reduce_sum(acc);
  __shared__ float wsum[8];
  if ((tid & 31) == 0) wsum[tid >> 5] = s;
  __syncthreads();
  if (tid == 0) {
    float tot = 0.0f;
    for (int w = 0; w < 8; ++w) tot += wsum[w];
    blkNeg[b * 32 + blk] = tot;
  }
}

__global__ __launch_bounds__(256) void k_finalize(char* __restrict__ ws,
                                                  float* __restrict__ out) {
  uint32_t* u = (uint32_t*)(ws + OFF_U32);
  const float* blkLoc = (const float*)(u + W_BLKLOC);
  const float* blkPC = (const float*)(u + W_BLKPC);
  const float* blkNeg = (const float*)(u + W_BLKNEG);
  const uint32_t* posCnt = u + W_POSCNT;
  const uint32_t* kTgt = u + W_KTGT;
  const uint32_t* cntGt = u + W_CNTGT;
  const uint32_t* tBits = u + W_TBITS;

  __shared__ float sLoc[256], sPC[256], sNeg[256];
  const int tid = threadIdx.x;
  float accL = 0.0f, accP = 0.0f;
  for (int i = tid; i < B_ * 128; i += 256) { accL += blkLoc[i]; accP += blkPC[i]; }
  sLoc[tid] = accL;
  sPC[tid] = accP;
  float ng = 0.0f;
  if (tid < B_) {
    uint32_t k = kTgt[tid];
    if (k > 0u) {
      float s = 0.0f;
      for (int j = 0; j < 32; ++j) s += blkNeg[tid * 32 + j];
      float tv = __uint_as_float(tBits[tid]);
      ng = s + (float)(k - cntGt[tid]) * tv;  // ties at threshold
    }
  }
  sNeg[tid] = ng;
  __syncthreads();
  for (int off = 128; off > 0; off >>= 1) {
    if (tid < off) {
      sLoc[tid] += sLoc[tid + off];
      sPC[tid] += sPC[tid + off];
      sNeg[tid] += sNeg[tid + off];
    }
    __syncthreads();
  }
  if (tid == 0) {
    uint32_t pc = 0;
    for (int bb = 0; bb < B_; ++bb) pc += posCnt[bb];
    float posSum = fmaxf((float)pc, 1.0f);
    out[0] = sLoc[0] / posSum;
    out[1] = 1.0f /*CONF_GAIN*/ * (sPC[0] + sNeg[0]) / posSum;
  }
}

// ---------------- launch ----------------
extern "C" void kernel_launch(void* const* d_in, const int* in_sizes, int n_in,
                              void* d_out, int out_size, void* d_ws, size_t ws_size,
                              hipStream_t stream) {
  (void)in_sizes; (void)n_in; (void)out_size; (void)ws_size;
  const float* loc_p = (const float*)d_in[0];
  const float* conf_p = (const float*)d_in[1];
  const float* priors = (const float*)d_in[2];
  const float* targets = (const float*)d_in[3];
  float* out = (float*)d_out;
  char* ws = (char*)d_ws;

  k_init<<<1024, 256, 0, stream>>>(ws);
  k_best_prior<<<B_ * 64, 256, 0, stream>>>(priors, targets, ws);
  k_main<<<B_ * 128, 256, 0, stream>>>(loc_p, conf_p, priors, targets, ws);
  k_select<<<B_, 32, 0, stream>>>(ws, 0);
  k_hist_refine<<<B_ * 32, 256, 0, stream>>>(ws, 1);
  k_select<<<B_, 32, 0, stream>>>(ws, 1);
  k_hist_refine<<<B_ * 32, 256, 0, stream>>>(ws, 2);
  k_select<<<B_, 32, 0, stream>>>(ws, 2);
  k_sum_gt<<<B_ * 32, 256, 0, stream>>>(ws);
  k_finalize<<<1, 256, 0, stream>>>(ws, out);
}